// CrossAttention_180388626454
// MI455X (gfx1250) — compile-verified
//
#include <hip/hip_runtime.h>

// Problem constants (from reference): B, NQ, NV, DIM, H, F = 32,197,1568,768,12,8
#define B_    32
#define NQ_   197
#define NV_   1568
#define DIM_  768
#define H_    12
#define F_    8
#define D_    64
#define NPF   196           // keys per fold == queries per fold (NQ-1)
#define NOUT  1569          // 1 cls + F*NPF
#define KVW   (2 * DIM_)    // 1536

#define AS1 __attribute__((address_space(1)))
#define AS3 __attribute__((address_space(3)))

typedef __attribute__((ext_vector_type(2))) float v2f;
typedef __attribute__((ext_vector_type(8))) float v8f;
typedef int v4i_vs __attribute__((vector_size(16)));   // <4 x i32>, matches builtin

static __device__ __forceinline__ v8f wmma4(v2f a, v2f b, v8f c) {
  // D(16x16,f32) = A(16x4,f32) x B(4x16,f32) + C
  return __builtin_amdgcn_wmma_f32_16x16x4_f32(false, a, false, b, (short)0, c,
                                               false, false);
}

// Async DMA of 16 bytes global -> LDS (GLOBAL_LOAD_ASYNC_TO_LDS_B128).
// Tracked by ASYNCcnt; both addresses must be 16B-aligned.
static __device__ __forceinline__ void async_copy16(const float* gp, float* lp) {
  __builtin_amdgcn_global_load_async_to_lds_b128(
      (AS1 v4i_vs*)gp, (AS3 v4i_vs*)lp, 0, 0);
}

// ---------------------------------------------------------------------------
// Generic fp32 WMMA GEMM: C[M,N] = alpha * (A[M,K] @ B[K,N]) + bias[N]
// Wave computes a 32x32 tile (2x2 of 16x16 accumulators), K-step 4.
// Block: 256 threads = 8 waves arranged 4(M) x 2(N) -> 128x64 block tile.
// N must be a multiple of 64; M is bounds-checked; K multiple of 4.
// ---------------------------------------------------------------------------
__global__ __launch_bounds__(256) void gemm_wmma_f32(
    const float* __restrict__ A, const float* __restrict__ Bm,
    float* __restrict__ C, const float* __restrict__ bias,
    int M, int N, int K, float alpha)
{
  const int lane = threadIdx.x & 31;
  const int wave = threadIdx.x >> 5;
  const int lh = lane & 15;     // 0..15
  const int lg = lane >> 4;     // 0 or 1 (selects K pair / M+8 rows)

  const int m0 = blockIdx.y * 128 + (wave >> 1) * 32;
  const int n0 = blockIdx.x * 64  + (wave & 1) * 32;

  int ra0 = m0 + lh;      if (ra0 >= M) ra0 = M - 1;
  int ra1 = m0 + 16 + lh; if (ra1 >= M) ra1 = M - 1;
  const float* Ap0 = A + (size_t)ra0 * K + 2 * lg;   // A-frag: lanes>=16 hold K+2,K+3
  const float* Ap1 = A + (size_t)ra1 * K + 2 * lg;
  const float* Bp  = Bm + (size_t)(2 * lg) * N;      // B-frag: lanes>=16 hold rows K+2,K+3
  const int cb0 = n0 + lh, cb1 = n0 + 16 + lh;

  v8f acc00 = {}, acc01 = {}, acc10 = {}, acc11 = {};
  for (int k0 = 0; k0 < K; k0 += 4) {
    v2f a0 = *(const v2f*)(Ap0 + k0);
    v2f a1 = *(const v2f*)(Ap1 + k0);
    const float* Bk = Bp + (size_t)k0 * N;
    v2f b0, b1;
    b0.x = Bk[cb0]; b0.y = Bk[cb0 + N];
    b1.x = Bk[cb1]; b1.y = Bk[cb1 + N];
    acc00 = wmma4(a0, b0, acc00);
    acc01 = wmma4(a0, b1, acc01);
    acc10 = wmma4(a1, b0, acc10);
    acc11 = wmma4(a1, b1, acc11);
  }

  auto st = [&](const v8f& acc, int mt, int nt) {
    const int col = nt + lh;
    const float bv = bias ? bias[col] : 0.0f;
#pragma unroll
    for (int r = 0; r < 8; ++r) {
      const int m = mt + r + 8 * lg;   // C-layout: VGPR r = row r (lanes 0-15) / r+8 (16-31)
      if (m < M) C[(size_t)m * N + col] = acc[r] * alpha + bv;
    }
  };
  st(acc00, m0, n0);      st(acc01, m0, n0 + 16);
  st(acc10, m0 + 16, n0); st(acc11, m0 + 16, n0 + 16);
}

// ---------------------------------------------------------------------------
// Fold attention: one block per (b, h, f). 13 waves x 16 query rows = 208 (196
// valid). Flash-style online softmax over 13 key tiles of 16. K/V tiles are
// DMA'd into LDS with GLOBAL_LOAD_ASYNC_TO_LDS_B128 (ASYNCcnt + barrier),
// LDS row stride 68 floats -> conflict-free column reads. P goes through a
// per-wave padded LDS tile to convert C-layout -> A-layout for the P@V WMMA.
// ---------------------------------------------------------------------------
__global__ __launch_bounds__(416) void attn_fold_kernel(
    const float* __restrict__ Qb,   // (B, NQ, DIM), already scaled by d^-0.5
    const float* __restrict__ KVb,  // (B, NV, 2*DIM): [:,:,0:768]=K, [768:]=V
    float* __restrict__ Ctx)        // (B, NOUT, DIM)
{
  __shared__ float sK[16 * 68];
  __shared__ float sV[16 * 68];
  __shared__ float sP[13][16 * 17];

  const int idx = blockIdx.x;
  const int f = idx % F_;
  const int h = (idx / F_) % H_;
  const int b = idx / (F_ * H_);

  const int lane = threadIdx.x & 31;
  const int wave = threadIdx.x >> 5;   // 0..12
  const int lh = lane & 15;
  const int lg = lane >> 4;

  // Load this wave's 16x64 Q tile as 16 A-fragments (K-step 4).
  const int qrow = wave * 16 + lh;
  const int qr = qrow < NPF ? qrow : NPF - 1;          // clamp padded rows
  const float* qp = Qb + ((size_t)b * NQ_ + 1 + qr) * DIM_ + h * D_ + 2 * lg;
  v2f qf[16];
#pragma unroll
  for (int s = 0; s < 16; ++s) qf[s] = *(const v2f*)(qp + 4 * s);

  v8f o0 = {}, o1 = {}, o2 = {}, o3 = {};
  float runM[8], runL[8];
#pragma unroll
  for (int r = 0; r < 8; ++r) { runM[r] = -1e30f; runL[r] = 0.0f; }

  const float* Kb = KVb + (size_t)b * NV_ * KVW + h * D_;
  const float* Vb = Kb + DIM_;
  float* Pw = &sP[wave][0];

  for (int j = 0; j < 13; ++j) {
    __syncthreads();   // previous tile fully consumed before overwrite
    // Async-DMA K_j / V_j tiles (16 keys x 64 dims each) into LDS in 16B
    // chunks: 2 * 256 transfers spread over 416 lanes. Out-of-range keys are
    // clamp-loaded; their scores are masked to -inf below, so P==0 for them.
    for (int u = threadIdx.x; u < 512; u += 416) {
      const int mat = u >> 8;            // 0 = K, 1 = V
      const int e = u & 255;
      const int row = e >> 4;            // 0..15  (key within tile)
      const int c4 = (e & 15) << 2;      // 0,4,...,60 (dim group of 4 floats)
      int kk = j * 16 + row;
      if (kk >= NPF) kk = NPF - 1;
      const float* gp = (mat ? Vb : Kb) + (size_t)(f * NPF + kk) * KVW + c4;
      float* lp = (mat ? sV : sK) + row * 68 + c4;
      async_copy16(gp, lp);
    }
    asm volatile("s_wait_asynccnt 0x0" ::: "memory");
    __syncthreads();

    // S(16x16) = Q(16x64) . K_j(16x64)^T  -> 16 WMMAs
    v8f S = {};
#pragma unroll
    for (int s = 0; s < 16; ++s) {
      const int k0 = 4 * s + 2 * lg;
      v2f bk = *(const v2f*)&sK[lh * 68 + k0];   // B[k][n] = K[n][k]
      S = wmma4(qf[s], bk, S);
    }
    // Mask padded keys (only tile j==12 has them); key index per lane = j*16+lh.
    if (j * 16 + lh >= NPF) {
#pragma unroll
      for (int r = 0; r < 8; ++r) S[r] = -1e30f;
    }

    // Online softmax: row r+8*lg lives across the lane's 16-lane half.
#pragma unroll
    for (int r = 0; r < 8; ++r) {
      float v = S[r];
      v = fmaxf(v, __shfl_xor(v, 1, 32));
      v = fmaxf(v, __shfl_xor(v, 2, 32));
      v = fmaxf(v, __shfl_xor(v, 4, 32));
      v = fmaxf(v, __shfl_xor(v, 8, 32));
      const float nm = fmaxf(runM[r], v);
      const float sc = __expf(runM[r] - nm);
      runM[r] = nm;
      float p = __expf(S[r] - nm);
      S[r] = p;
      float ps = p;
      ps += __shfl_xor(ps, 1, 32);
      ps += __shfl_xor(ps, 2, 32);
      ps += __shfl_xor(ps, 4, 32);
      ps += __shfl_xor(ps, 8, 32);
      runL[r] = runL[r] * sc + ps;
      o0[r] *= sc; o1[r] *= sc; o2[r] *= sc; o3[r] *= sc;
    }

    // C-layout -> A-layout via per-wave LDS tile.
#pragma unroll
    for (int r = 0; r < 8; ++r) Pw[(r + 8 * lg) * 17 + lh] = S[r];
    asm volatile("s_wait_dscnt 0x0" ::: "memory");

    // O(16x64) += P(16x16) . V_j(16x64)  -> 16 WMMAs
#pragma unroll
    for (int s = 0; s < 4; ++s) {
      const int k0 = 4 * s + 2 * lg;
      v2f pa; pa.x = Pw[lh * 17 + k0]; pa.y = Pw[lh * 17 + k0 + 1];
      v2f bv;
      bv.x = sV[k0 * 68 + lh];       bv.y = sV[(k0 + 1) * 68 + lh];
      o0 = wmma4(pa, bv, o0);
      bv.x = sV[k0 * 68 + 16 + lh];  bv.y = sV[(k0 + 1) * 68 + 16 + lh];
      o1 = wmma4(pa, bv, o1);
      bv.x = sV[k0 * 68 + 32 + lh];  bv.y = sV[(k0 + 1) * 68 + 32 + lh];
      o2 = wmma4(pa, bv, o2);
      bv.x = sV[k0 * 68 + 48 + lh];  bv.y = sV[(k0 + 1) * 68 + 48 + lh];
      o3 = wmma4(pa, bv, o3);
    }
  }

  // Epilogue: normalize and scatter into (B, NOUT, DIM) at col h*64.
#pragma unroll
  for (int r = 0; r < 8; ++r) {
    const int m = wave * 16 + r + 8 * lg;
    if (m < NPF) {
      const float inv = 1.0f / runL[r];
      float* outp =
          Ctx + ((size_t)b * NOUT + 1 + (size_t)f * NPF + m) * DIM_ + h * D_;
      outp[lh]      = o0[r] * inv;
      outp[16 + lh] = o1[r] * inv;
      outp[32 + lh] = o2[r] * inv;
      outp[48 + lh] = o3[r] * inv;
    }
  }
}

// ---------------------------------------------------------------------------
// CLS attention: one block per (b,h). cls_q(1x64) vs all 1568 keys. Tiny.
// ---------------------------------------------------------------------------
__global__ __launch_bounds__(256) void attn_cls_kernel(
    const float* __restrict__ Qb, const float* __restrict__ KVb,
    float* __restrict__ Ctx)
{
  __shared__ float sq[64];
  __shared__ float sc_[NV_];
  __shared__ float red[256];

  const int bh = blockIdx.x;
  const int h = bh % H_;
  const int b = bh / H_;
  const int tid = threadIdx.x;

  const float* qp = Qb + (size_t)b * NQ_ * DIM_ + h * D_;
  if (tid < 64) sq[tid] = qp[tid];
  __syncthreads();

  const float* Kb = KVb + (size_t)b * NV_ * KVW + h * D_;
  for (int key = tid; key < NV_; key += 256) {
    const float* kr = Kb + (size_t)key * KVW;
    float s = 0.0f;
#pragma unroll 8
    for (int d2 = 0; d2 < 64; ++d2) s += sq[d2] * kr[d2];
    sc_[key] = s;
  }
  __syncthreads();

  float m = -1e30f;
  for (int key = tid; key < NV_; key += 256) m = fmaxf(m, sc_[key]);
  red[tid] = m;
  __syncthreads();
  for (int s2 = 128; s2 > 0; s2 >>= 1) {
    if (tid < s2) red[tid] = fmaxf(red[tid], red[tid + s2]);
    __syncthreads();
  }
  m = red[0];
  __syncthreads();

  float lsum = 0.0f;
  for (int key = tid; key < NV_; key += 256) {
    const float p = __expf(sc_[key] - m);
    sc_[key] = p;
    lsum += p;
  }
  __syncthreads();
  red[tid] = lsum;
  __syncthreads();
  for (int s2 = 128; s2 > 0; s2 >>= 1) {
    if (tid < s2) red[tid] += red[tid + s2];
    __syncthreads();
  }
  const float invL = 1.0f / red[0];
  __syncthreads();

  if (tid < 64) {
    const float* Vb = Kb + DIM_;
    float acc = 0.0f;
    for (int key = 0; key < NV_; ++key)
      acc += sc_[key] * Vb[(size_t)key * KVW + tid];
    Ctx[(size_t)b * NOUT * DIM_ + h * D_ + tid] = acc * invL;
  }
}

// ---------------------------------------------------------------------------
extern "C" void kernel_launch(void* const* d_in, const int* in_sizes, int n_in,
                              void* d_out, int out_size, void* d_ws,
                              size_t ws_size, hipStream_t stream)
{
  const float* x        = (const float*)d_in[0];  // (B, NV, DIM)
  const float* question = (const float*)d_in[1];  // (B, NQ, DIM)
  const float* Wq       = (const float*)d_in[2];  // (DIM, DIM)
  const float* Wkv      = (const float*)d_in[3];  // (DIM, 2*DIM)
  const float* Wproj    = (const float*)d_in[4];  // (DIM, DIM)
  const float* bproj    = (const float*)d_in[5];  // (DIM,)
  (void)in_sizes; (void)n_in; (void)out_size; (void)ws_size;

  // Workspace layout (floats): qbuf | kv | ctx   (~482 MB total)
  float* qbuf = (float*)d_ws;                          // (B, NQ, DIM)
  float* kv   = qbuf + (size_t)B_ * NQ_ * DIM_;        // (B, NV, 2*DIM)
  float* ctx  = kv   + (size_t)B_ * NV_ * KVW;         // (B, NOUT, DIM)

  const int Mkv = B_ * NV_;    // 50176
  const int Mq  = B_ * NQ_;    // 6304
  const int Mo  = B_ * NOUT;   // 50208

  // kv = x @ Wkv
  gemm_wmma_f32<<<dim3(KVW / 64, (Mkv + 127) / 128), 256, 0, stream>>>(
      x, Wkv, kv, nullptr, Mkv, KVW, DIM_, 1.0f);
  // q = (question @ Wq) * d^-0.5
  gemm_wmma_f32<<<dim3(DIM_ / 64, (Mq + 127) / 128), 256, 0, stream>>>(
      question, Wq, qbuf, nullptr, Mq, DIM_, DIM_, 0.125f);
  // fold attention + cls attention -> ctx
  attn_fold_kernel<<<B_ * H_ * F_, 416, 0, stream>>>(qbuf, kv, ctx);
  attn_cls_kernel<<<B_ * H_, 256, 0, stream>>>(qbuf, kv, ctx);
  // out = ctx @ Wproj + bproj
  gemm_wmma_f32<<<dim3(DIM_ / 64, (Mo + 127) / 128), 256, 0, stream>>>(
      ctx, Wproj, (float*)d_out, bproj, Mo, DIM_, DIM_, 1.0f);
}